// _AnchorLayer_36249523978388
// MI455X (gfx1250) — compile-verified
//
#include <hip/hip_runtime.h>

namespace {
constexpr int B = 8, A = 9, H = 128, W = 128, K = 64;
constexpr int N = A * H * W;     // 147456 anchors
constexpr int BK = B * K;        // 512 gt entries
constexpr int BLOCK = 256;       // N == 576 * 256 exactly
constexpr float POS_OV = 0.7f;
constexpr float NEG_OV = 0.3f;
constexpr float INV_SCALE = 1.0f / 16.0f;  // 1/(IMAGE_DIM/H); power of 2 -> exact
}

#if defined(__has_builtin)
#if __has_builtin(__builtin_amdgcn_global_load_async_to_lds_b128) && \
    __has_builtin(__builtin_amdgcn_s_wait_asynccnt)
#define ANCHOR_ASYNC_LDS 1
#endif
#endif

#if ANCHOR_ASYNC_LDS
typedef int v4i_vec __attribute__((__vector_size__(4 * sizeof(int))));
typedef __attribute__((address_space(1))) v4i_vec* as1_v4i;
typedef __attribute__((address_space(3))) v4i_vec* as3_v4i;
#endif

__global__ __launch_bounds__(BLOCK) void _AnchorLayer_kernel(
    const float* __restrict__ anchors,   // (N,4)
    const float* __restrict__ gt,        // (B,K,4)
    float* __restrict__ labels,          // (B,N)
    float* __restrict__ targets)         // (B,N,4)
{
    __shared__ float4 s_raw[BK];   // staged raw gt boxes (8 KB)
    __shared__ float4 s_box[BK];   // x/16, y/16, (x+w)/16, (y+h)/16
    __shared__ float  s_area[BK];  // (w/16)*(h/16)
    __shared__ float2 s_wh[BK];    // w/16, h/16

    const int tid = threadIdx.x;
    const int n   = blockIdx.x * BLOCK + tid;

    // Kick the anchor cacheline fetch early (emits global_prefetch_b8).
    __builtin_prefetch(anchors + (size_t)n * 4, 0, 3);

    // ---- Stage raw gt boxes (512 x float4) into LDS ----
#if ANCHOR_ASYNC_LDS
    for (int e = tid; e < BK; e += BLOCK) {
        __builtin_amdgcn_global_load_async_to_lds_b128(
            (as1_v4i)(uintptr_t)(gt + (size_t)e * 4),
            (as3_v4i)(uint32_t)(uintptr_t)(&s_raw[e]),
            0, 0);
    }
    __builtin_amdgcn_s_wait_asynccnt(0);
#else
    for (int e = tid; e < BK; e += BLOCK) {
        s_raw[e] = ((const float4*)gt)[e];
    }
#endif
    __syncthreads();

    // ---- Derive per-gt quantities, pre-scaled by 1/16 (bit-exact vs ref) ----
    for (int e = tid; e < BK; e += BLOCK) {
        float4 g = s_raw[e];
        float gx = g.x * INV_SCALE;
        float gy = g.y * INV_SCALE;
        float gw = g.z * INV_SCALE;
        float gh = g.w * INV_SCALE;
        s_box[e]  = make_float4(gx, gy, gx + gw, gy + gh);
        s_area[e] = gw * gh;
        s_wh[e]   = make_float2(gw, gh);
    }
    __syncthreads();

    // ---- Per-anchor state (hoisted across all 8 batches) ----
    const float4 a = ((const float4*)anchors)[n];
    const float ax = a.x, ay = a.y, aw = a.z, ah = a.w;
    const float ax2 = ax + aw, ay2 = ay + ah, aarea = aw * ah;
    const bool valid = (ax >= 0.0f) && (ay >= 0.0f) &&
                       (ax2 < (float)H) && (ay2 < (float)W);

    for (int b = 0; b < B; ++b) {
        const int base = b * K;
        float bi = 0.0f;   // best intersection
        float bu = 1.0f;   // best union (positive)
        int bidx = 0;
        #pragma unroll 16
        for (int k = 0; k < K; ++k) {
            const float4 q = s_box[base + k];   // LDS broadcast (uniform addr)
            const float qa = s_area[base + k];
            float iw = fminf(ax2, q.z) - fmaxf(ax, q.x);
            float ih = fminf(ay2, q.w) - fmaxf(ay, q.y);
            iw = fmaxf(iw, 0.0f);
            ih = fmaxf(ih, 0.0f);
            const float inter = iw * ih;
            const float uni   = aarea + qa - inter;   // >= max(areas) > 0
            // iou_k > iou_best  <=>  inter*bu > bi*uni  (division-free argmax,
            // strict '>' keeps first-max semantics like jnp.argmax)
            const bool upd = inter * bu > bi * uni;
            bi   = upd ? inter : bi;
            bu   = upd ? uni   : bu;
            bidx = upd ? k     : bidx;
        }
        const float maxov = bi / bu;   // single divide per (b, anchor)
        float lab = -1.0f;
        lab = (maxov >= POS_OV) ? 1.0f : lab;
        lab = (maxov <= NEG_OV) ? 0.0f : lab;
        lab = valid ? lab : -1.0f;

        const float4 q  = s_box[base + bidx];
        const float2 wh = s_wh[base + bidx];
        float4 t;
        t.x = valid ? (ax - q.x)  : 0.0f;
        t.y = valid ? (ay - q.y)  : 0.0f;
        t.z = valid ? (aw - wh.x) : 0.0f;
        t.w = valid ? (ah - wh.y) : 0.0f;

        labels[(size_t)b * N + n] = lab;
        ((float4*)targets)[(size_t)b * N + n] = t;
    }
}

extern "C" void kernel_launch(void* const* d_in, const int* in_sizes, int n_in,
                              void* d_out, int out_size, void* d_ws, size_t ws_size,
                              hipStream_t stream) {
    // d_in[0] = cls_scores (unused by the reference output)
    const float* anchors = (const float*)d_in[1];   // (A,H,W,4) -> (N,4)
    const float* gt      = (const float*)d_in[2];   // (B,K,4)
    float* labels  = (float*)d_out;                       // B*N floats
    float* targets = (float*)d_out + (size_t)B * N;       // B*N*4 floats
    _AnchorLayer_kernel<<<N / BLOCK, BLOCK, 0, stream>>>(anchors, gt, labels, targets);
}